// Graph_QMixer_25503515803797
// MI455X (gfx1250) — compile-verified
//
#include <hip/hip_runtime.h>

#define NAG 16
#define OBSF 128
#define STATEF 256
#define EMBEDF 32
#define NHID 32
#define NHEADS 4
#define ALPHA 0.2f
#define NEGV (-9.0e15f)

// padded LDS strides (avoid 64-bank conflicts on row-strided reads)
#define OS 132   // obs / xcat row stride
#define WS 36    // per-head Wh row stride (16x32)
#define AS 20    // att row stride (16x16)
#define BS 36    // b_all row stride

typedef __attribute__((ext_vector_type(2))) float v2f;
typedef __attribute__((ext_vector_type(8))) float v8f;

__device__ __forceinline__ v8f wmma4(v2f a, v2f b, v8f c) {
  // D = A(16x4,f32) * B(4x16,f32) + C(16x16,f32)
  return __builtin_amdgcn_wmma_f32_16x16x4_f32(false, a, false, b, (short)0, c,
                                               false, false);
}

__device__ __forceinline__ float elu1(float x) { return x > 0.f ? x : (expf(x) - 1.f); }
__device__ __forceinline__ float lrelu(float x) { return x > 0.f ? x : ALPHA * x; }

// CDNA5 async copy: 16B global -> 16B LDS per lane, tracked on ASYNCcnt.
__device__ __forceinline__ void async_g2l_b128(unsigned lds_byte_addr,
                                               const void* gaddr) {
  asm volatile("global_load_async_to_lds_b128 %0, %1, off"
               :
               : "v"(lds_byte_addr), "v"((unsigned long long)(uintptr_t)gaddr)
               : "memory");
}
__device__ __forceinline__ void wait_async0() {
  asm volatile("s_wait_asynccnt 0x0" ::: "memory");
}
__device__ __forceinline__ unsigned lds_off(const void* p) {
  // flat LDS-aperture pointers carry the LDS byte offset in the low 32 bits
  return (unsigned)(uintptr_t)p;
}

// A fragment: A[m][k], 16x4 tile at k0. lanes 0-15: M=l16,K=k0+{0,1}; lanes 16-31: M=l16,K=k0+{2,3}
__device__ __forceinline__ v2f afrag(const float* base, int row, int rs, int k0, int half) {
  const float* p = base + row * rs + k0 + 2 * half;
  v2f a = {p[0], p[1]};
  return a;
}
// B fragment: B[k][n], 4x16 tile at (k0, col0+l16); mirrors A's K split across lane halves
__device__ __forceinline__ v2f bfrag(const float* base, int k0, int ks, int col, int half) {
  const float* p = base + (k0 + 2 * half) * ks + col;
  v2f b = {p[0], p[ks]};
  return b;
}

// softmax over axis=1 (rows i) per column j of e[i][j]=leaky(f1[i]+f2[j]), masked by adj
__device__ __forceinline__ void col_softmax(float* s_att, const float* s_f,
                                            const float* s_adj, int lane) {
  if (lane < 16) {
    const int j = lane;
    const float f2j = s_f[16 + j];
    float ev[16];
    float m = -3.0e38f;
#pragma unroll
    for (int i = 0; i < 16; ++i) {
      float e = lrelu(s_f[i] + f2j);
      e = (s_adj[i * 16 + j] > 0.f) ? e : NEGV;
      ev[i] = e;
      m = fmaxf(m, e);
    }
    float s = 0.f;
#pragma unroll
    for (int i = 0; i < 16; ++i) { ev[i] = expf(ev[i] - m); s += ev[i]; }
    const float inv = 1.f / s;
#pragma unroll
    for (int i = 0; i < 16; ++i) s_att[i * AS + j] = ev[i] * inv;
  }
}

// 4 GAT heads: obs[16,128] -> elu(att@Wh) concat into s_xcat[16,128]
__device__ __forceinline__ void run_heads(const float* __restrict__ Wh_g,
                                          const float* __restrict__ ah_g,
                                          float* s_xcat, float* s_wh, float* s_att,
                                          float* s_f, const float* s_obs,
                                          const float* s_adj, int lane, int half, int l16) {
  for (int h = 0; h < NHEADS; ++h) {
    const float* W = Wh_g + h * OBSF * NHID;   // [128][32] row-major
    const float* ah = ah_g + h * 2 * NHID;     // [64]
    v8f c0 = {}; v8f c1 = {};
    for (int k0 = 0; k0 < OBSF; k0 += 4) {
      v2f a = afrag(s_obs, l16, OS, k0, half);
      c0 = wmma4(a, bfrag(W, k0, NHID, l16, half), c0);
      c1 = wmma4(a, bfrag(W, k0, NHID, 16 + l16, half), c1);
    }
#pragma unroll
    for (int r = 0; r < 8; ++r) {
      const int row = r + 8 * half;
      s_wh[row * WS + l16] = c0[r];
      s_wh[row * WS + 16 + l16] = c1[r];
    }
    __syncthreads();
    {  // f1 (lanes 0-15) / f2 (lanes 16-31): Wh @ a[:32] / Wh @ a[32:]
      const float* av = ah + half * NHID;
      float acc = 0.f;
#pragma unroll
      for (int f = 0; f < NHID; ++f) acc += s_wh[l16 * WS + f] * av[f];
      s_f[lane] = acc;
    }
    __syncthreads();
    col_softmax(s_att, s_f, s_adj, lane);
    __syncthreads();
    v8f d0 = {}; v8f d1 = {};
#pragma unroll
    for (int k0 = 0; k0 < 16; k0 += 4) {
      v2f a = afrag(s_att, l16, AS, k0, half);
      d0 = wmma4(a, bfrag(s_wh, k0, WS, l16, half), d0);
      d1 = wmma4(a, bfrag(s_wh, k0, WS, 16 + l16, half), d1);
    }
#pragma unroll
    for (int r = 0; r < 8; ++r) {
      const int row = r + 8 * half;
      s_xcat[row * OS + h * 32 + l16] = elu1(d0[r]);
      s_xcat[row * OS + h * 32 + 16 + l16] = elu1(d1[r]);
    }
    __syncthreads();
  }
}

// out layer prep: Wh2 = xcat @ Wout into s_wh2[16][FS]; att2 into s_att
template <int F, int FS>
__device__ __forceinline__ void gat_out_prep(const float* __restrict__ Wout,
                                             const float* __restrict__ aout,
                                             const float* s_xcat, float* s_wh2,
                                             float* s_att, float* s_f, const float* s_adj,
                                             int lane, int half, int l16) {
  for (int nt = 0; nt < F / 16; ++nt) {
    v8f c = {};
    for (int k0 = 0; k0 < OBSF; k0 += 4) {
      v2f a = afrag(s_xcat, l16, OS, k0, half);
      c = wmma4(a, bfrag(Wout, k0, F, nt * 16 + l16, half), c);
    }
#pragma unroll
    for (int r = 0; r < 8; ++r) s_wh2[(r + 8 * half) * FS + nt * 16 + l16] = c[r];
  }
  __syncthreads();
  {
    const float* av = aout + half * F;
    float acc = 0.f;
    for (int f = 0; f < F; ++f) acc += s_wh2[l16 * FS + f] * av[f];
    s_f[lane] = acc;
  }
  __syncthreads();
  col_softmax(s_att, s_f, s_adj, lane);
  __syncthreads();
}

// one 16-col output tile: att2 @ Wh2, then elu -> column log_softmax (over 16 agent rows,
// split across lane halves in D layout) -> abs. Result stays in D register layout.
template <int FS>
__device__ __forceinline__ v8f out_tile(const float* s_att, const float* s_wh2, int nt,
                                        int half, int l16) {
  v8f d = {};
#pragma unroll
  for (int k0 = 0; k0 < 16; k0 += 4) {
    v2f a = afrag(s_att, l16, AS, k0, half);
    d = wmma4(a, bfrag(s_wh2, k0, FS, nt * 16 + l16, half), d);
  }
#pragma unroll
  for (int r = 0; r < 8; ++r) d[r] = elu1(d[r]);
  float lmax = d[0];
#pragma unroll
  for (int r = 1; r < 8; ++r) lmax = fmaxf(lmax, d[r]);
  const float cmax = fmaxf(lmax, __shfl_xor(lmax, 16, 32));
  float lsum = 0.f;
#pragma unroll
  for (int r = 0; r < 8; ++r) lsum += expf(d[r] - cmax);
  const float csum = lsum + __shfl_xor(lsum, 16, 32);
  const float lg = logf(csum);
#pragma unroll
  for (int r = 0; r < 8; ++r) d[r] = fabsf(d[r] - cmax - lg);
  return d;
}

__global__ __launch_bounds__(32) void qmixer_kernel(
    const float* __restrict__ agent_qs, const float* __restrict__ states,
    const float* __restrict__ obs_ls, const float* __restrict__ adj_ls,
    const float* __restrict__ wn_w, const float* __restrict__ wn_b,
    const float* __restrict__ g1_Wh, const float* __restrict__ g1_ah,
    const float* __restrict__ g1_Wout, const float* __restrict__ g1_aout,
    const float* __restrict__ gf_Wh, const float* __restrict__ gf_ah,
    const float* __restrict__ gf_Wout, const float* __restrict__ gf_aout,
    const float* __restrict__ hb_W, const float* __restrict__ hb_b,
    const float* __restrict__ v1_w, const float* __restrict__ v1_b,
    const float* __restrict__ v2_w, const float* __restrict__ v2_b,
    float* __restrict__ out, int B) {
  const int b = blockIdx.x;
  if (b >= B) return;
  const int lane = threadIdx.x;
  const int half = lane >> 4, l16 = lane & 15;

  __shared__ float s_obs[16 * OS];
  __shared__ float s_xcat[16 * OS];
  __shared__ float s_wh2[16 * 516];  // out-layer Wh2 (g1: 16x512)
  __shared__ float s_wh[16 * WS];
  __shared__ float s_att[16 * AS];
  __shared__ float s_adj[256];
  __shared__ float s_st[256];
  __shared__ float s_ball[16 * BS];
  __shared__ float s_f[32];
  __shared__ float s_y[16];
  __shared__ float s_dis[16];

  // ---- stage per-batch inputs with CDNA5 async global->LDS B128 copies ----
  {
    const float* obs = obs_ls + (size_t)b * NAG * OBSF;  // 2048 floats, 512 x 16B chunks
    const unsigned obs_base = lds_off(s_obs);
#pragma unroll
    for (int t = 0; t < 16; ++t) {
      const int c = t * 32 + lane;             // chunk id; 32 chunks per 128-float row
      const int row = c >> 5, col = (c & 31) * 4;
      async_g2l_b128(obs_base + (unsigned)(row * OS + col) * 4u, obs + c * 4);
    }
    const float* adj = adj_ls + (size_t)b * NAG * NAG;   // 256 floats = 64 chunks
    const unsigned adj_base = lds_off(s_adj);
#pragma unroll
    for (int t = 0; t < 2; ++t) {
      const int c = t * 32 + lane;
      async_g2l_b128(adj_base + (unsigned)(c * 16), adj + c * 4);
    }
    const float* st = states + (size_t)b * STATEF;       // 256 floats = 64 chunks
    const unsigned st_base = lds_off(s_st);
#pragma unroll
    for (int t = 0; t < 2; ++t) {
      const int c = t * 32 + lane;
      async_g2l_b128(st_base + (unsigned)(c * 16), st + c * 4);
    }
    wait_async0();
  }
  __syncthreads();

  // ---- GAT #1 (hyper_w_1): fold |log_softmax(elu(out))| tiles directly into hidden acc ----
  run_heads(g1_Wh, g1_ah, s_xcat, s_wh, s_att, s_f, s_obs, s_adj, lane, half, l16);
  gat_out_prep<512, 516>(g1_Wout, g1_aout, s_xcat, s_wh2, s_att, s_f, s_adj, lane, half, l16);

  v8f hidA = {}; v8f hidB = {};  // hidden[i][e] pre-bias, D layout: e=l16 / e=16+l16
  {
    const float* qs = agent_qs + (size_t)b * NAG;
    for (int nt = 0; nt < 32; ++nt) {
      v8f w = out_tile<516>(s_att, s_wh2, nt, half, l16);  // columns f=nt*16.. share n=nt/2
      const float qsn = qs[nt >> 1];
      if ((nt & 1) == 0) {
#pragma unroll
        for (int r = 0; r < 8; ++r) hidA[r] += qsn * w[r];
      } else {
#pragma unroll
        for (int r = 0; r < 8; ++r) hidB[r] += qsn * w[r];
      }
    }
  }
  __syncthreads();

  // ---- GAT #2 (hyper_w_final): result kept in registers (D layout) ----
  run_heads(gf_Wh, gf_ah, s_xcat, s_wh, s_att, s_f, s_obs, s_adj, lane, half, l16);
  gat_out_prep<32, 36>(gf_Wout, gf_aout, s_xcat, s_wh2, s_att, s_f, s_adj, lane, half, l16);
  v8f wfA = out_tile<36>(s_att, s_wh2, 0, half, l16);
  v8f wfB = out_tile<36>(s_att, s_wh2, 1, half, l16);
  __syncthreads();

  // ---- b_all[i,e] = st . hb_W[i,e,:] + hb_b (VALU; weights L2-resident) ----
  for (int t = 0; t < 16; ++t) {
    const float* w = hb_W + ((size_t)t * 32 + lane) * STATEF;
    float acc = hb_b[t * 32 + lane];
    for (int s = 0; s < STATEF; ++s) acc += s_st[s] * w[s];
    s_ball[t * BS + lane] = acc;
  }
  // dis[j] = |st . wn_w[j] + wn_b[j]|
  if (lane < 16) {
    const float* w = wn_w + l16 * STATEF;
    float acc = wn_b[l16];
    for (int s = 0; s < STATEF; ++s) acc += s_st[s] * w[s];
    s_dis[l16] = fabsf(acc);
  }
  // V(st): relu(st@v1^T) @ v2^T + b2 ; one hidden unit per lane, warp-reduce
  float tv;
  {
    const float* w = v1_w + lane * STATEF;
    float acc = v1_b[lane];
    for (int s = 0; s < STATEF; ++s) acc += s_st[s] * w[s];
    tv = fmaxf(acc, 0.f) * v2_w[lane];
  }
#pragma unroll
  for (int m = 16; m >= 1; m >>= 1) tv += __shfl_xor(tv, m, 32);
  const float vval = tv + v2_b[0];
  __syncthreads();

  // ---- hidden = elu(acc + b_all); y[i] = sum_e hidden*wf; q = y.dis + v ----
  float p[8];
#pragma unroll
  for (int r = 0; r < 8; ++r) {
    const int row = r + 8 * half;
    const float hA = elu1(hidA[r] + s_ball[row * BS + l16]);
    const float hB = elu1(hidB[r] + s_ball[row * BS + 16 + l16]);
    p[r] = hA * wfA[r] + hB * wfB[r];
  }
#pragma unroll
  for (int r = 0; r < 8; ++r) {  // reduce over e (lanes within half)
    float x = p[r];
#pragma unroll
    for (int m = 8; m >= 1; m >>= 1) x += __shfl_xor(x, m, 32);
    p[r] = x;
  }
  if (l16 == 0) {
#pragma unroll
    for (int r = 0; r < 8; ++r) s_y[r + 8 * half] = p[r];
  }
  __syncthreads();
  if (lane == 0) {
    float q = vval;
#pragma unroll
    for (int j = 0; j < 16; ++j) q += s_y[j] * s_dis[j];
    out[b] = q;
  }
}

extern "C" void kernel_launch(void* const* d_in, const int* in_sizes, int n_in,
                              void* d_out, int out_size, void* d_ws, size_t ws_size,
                              hipStream_t stream) {
  (void)n_in; (void)out_size; (void)d_ws; (void)ws_size;
  const int B = in_sizes[0] / NAG;  // 64*128 = 8192
  qmixer_kernel<<<dim3(B), dim3(32), 0, stream>>>(
      (const float*)d_in[0], (const float*)d_in[1], (const float*)d_in[2],
      (const float*)d_in[3], (const float*)d_in[4], (const float*)d_in[5],
      (const float*)d_in[6], (const float*)d_in[7], (const float*)d_in[8],
      (const float*)d_in[9], (const float*)d_in[10], (const float*)d_in[11],
      (const float*)d_in[12], (const float*)d_in[13], (const float*)d_in[14],
      (const float*)d_in[15], (const float*)d_in[16], (const float*)d_in[17],
      (const float*)d_in[18], (const float*)d_in[19], (float*)d_out, B);
}